// HashManager_50165218017663
// MI455X (gfx1250) — compile-verified
//
#include <hip/hip_runtime.h>

// MI455X / gfx1250. Gather-bound multi-resolution voxel-hash interpolation.
// Levels: R = 16, 32, 64, 128; table rows = (R+1)^3 x float4.
// Level-0 table (17^3 = 4913 rows, 78,608 B) is staged into LDS per block,
// preferentially via gfx1250 async global->LDS loads (ASYNCcnt path).

typedef float v4f __attribute__((ext_vector_type(4)));
typedef int   v4i __attribute__((ext_vector_type(4)));

#define T0_ROWS 4913  // (16+1)^3

// One level: 8-corner gather + distance-weighted blend. Exactly mirrors the
// reference math: n = (R/2)*x + R/2; keys/dists computed on *scaled* corners.
__device__ __forceinline__ v4f interp(const v4f* __restrict__ table,
                                      float x, float y, float z, float R)
{
  const float h  = 0.5f * R;          // R/size with size==2
  const float R2 = R * R;
  const float nx = h * x + h;
  const float ny = h * y + h;
  const float nz = h * z + h;
  const float fx = floorf(nx), fy = floorf(ny), fz = floorf(nz);
  const float cx = ceilf(nx),  cy = ceilf(ny),  cz = ceilf(nz);

  int   idx[8];
  float d[8];
  float sumd = 0.0f;
#pragma unroll
  for (int b = 0; b < 8; ++b) {
    const float sx = ((b & 4) ? cx : fx);
    const float sy = ((b & 2) ? cy : fy) * R;
    const float sz = ((b & 1) ? cz : fz) * R2;
    // integer-valued f32 sums < 2^24 -> exact cast, matches reference
    idx[b] = (int)(sx + sy + sz);
    const float dx = sx - nx, dy = sy - ny, dz = sz - nz;
    d[b] = sqrtf(dx * dx + dy * dy + dz * dz);
    sumd += d[b];
  }
  const float rinv = 1.0f / sumd;
  v4f acc = {0.0f, 0.0f, 0.0f, 0.0f};
#pragma unroll
  for (int b = 0; b < 8; ++b) {
    const v4f e = table[idx[b]];      // global_load_b128 / ds_load_b128
    const float w = d[b] * rinv;
    acc += w * e;
  }
  return acc;
}

__global__ __launch_bounds__(256)
void hash_interp_kernel(const float* __restrict__ xyz,
                        const v4f* __restrict__ t0,
                        const v4f* __restrict__ t1,
                        const v4f* __restrict__ t2,
                        const v4f* __restrict__ t3,
                        float* __restrict__ out,
                        int n)
{
  __shared__ v4f lds0[T0_ROWS];       // 78,608 B of the 320 KB WGP LDS

  // ---- Stage level-0 table into LDS --------------------------------------
#if defined(__gfx1250__) && \
    __has_builtin(__builtin_amdgcn_global_load_async_to_lds_b128) && \
    __has_builtin(__builtin_amdgcn_s_wait_asynccnt)
  {
    // Signature (from clang diagnostic): param0 = AS1 ("__device__") v4i*,
    // param1 = AS3 (LDS) v4i*, then two imm ints (offset, cpol).
    typedef __attribute__((address_space(1))) v4i GV4;
    typedef __attribute__((address_space(3))) v4i LV4;
    for (int i = threadIdx.x; i < T0_ROWS; i += blockDim.x) {
      // per-lane 16B async copy: global_load_async_to_lds_b128 (ASYNCcnt)
      __builtin_amdgcn_global_load_async_to_lds_b128(
          (GV4*)(t0 + i), (LV4*)(lds0 + i), 0, 0);
    }
    __builtin_amdgcn_s_wait_asynccnt(0);
  }
#else
  for (int i = threadIdx.x; i < T0_ROWS; i += blockDim.x) lds0[i] = t0[i];
#endif
  __syncthreads();

  // ---- Grid-stride over points -------------------------------------------
  const int stride = gridDim.x * blockDim.x;
  for (int p = blockIdx.x * blockDim.x + threadIdx.x; p < n; p += stride) {
    // xyz is single-use streaming data: NT loads keep tables resident in L2
    const float x = __builtin_nontemporal_load(xyz + 3 * (size_t)p);
    const float y = __builtin_nontemporal_load(xyz + 3 * (size_t)p + 1);
    const float z = __builtin_nontemporal_load(xyz + 3 * (size_t)p + 2);

    const v4f o0 = interp(lds0, x, y, z, 16.0f);   // LDS gathers
    const v4f o1 = interp(t1,   x, y, z, 32.0f);   // L2-resident gathers
    const v4f o2 = interp(t2,   x, y, z, 64.0f);
    const v4f o3 = interp(t3,   x, y, z, 128.0f);

    v4f* outv = (v4f*)(out + (size_t)p * 16);
    __builtin_nontemporal_store(o0, outv + 0);     // 64 MB stream: TH_NT
    __builtin_nontemporal_store(o1, outv + 1);
    __builtin_nontemporal_store(o2, outv + 2);
    __builtin_nontemporal_store(o3, outv + 3);
  }
}

extern "C" void kernel_launch(void* const* d_in, const int* in_sizes, int n_in,
                              void* d_out, int out_size, void* d_ws, size_t ws_size,
                              hipStream_t stream)
{
  (void)n_in; (void)out_size; (void)d_ws; (void)ws_size;
  const float* xyz = (const float*)d_in[0];
  const v4f*   t0  = (const v4f*)d_in[1];
  const v4f*   t1  = (const v4f*)d_in[2];
  const v4f*   t2  = (const v4f*)d_in[3];
  const v4f*   t3  = (const v4f*)d_in[4];
  float*       out = (float*)d_out;

  const int n = in_sizes[0] / 3;      // N_POINTS
  dim3 block(256);                    // 8 wave32s; 78.6 KB LDS -> ~4 blocks/WGP
  dim3 grid(1024);                    // grid-stride: 4 points/thread

  hipLaunchKernelGGL(hash_interp_kernel, grid, block, 0, stream,
                     xyz, t0, t1, t2, t3, out, n);
}